// Diffusion_8993661518590
// MI455X (gfx1250) — compile-verified
//
#include <hip/hip_runtime.h>
#include <hip/hip_bf16.h>

// Problem constants (match reference)
#define LDIM 256       // latent size (one thread per l)
#define HDIM 16        // hidden width
#define BDIM 16384     // batch
#define NBLK 1024      // blocks
#define ROWS (BDIM / NBLK)   // 16 batch rows per block

// ---- CDNA5 async global->LDS staging (tracked by ASYNCcnt) --------------
// IOFFSET is applied to BOTH the LDS destination and the global source
// (CDNA5 ISA), so offset-congruent chunks need no address VALU.
__device__ __forceinline__ void async_ld_b128(unsigned lds_byte_addr, const void* gaddr, int ioff) {
    asm volatile("global_load_async_to_lds_b128 %0, %1, off offset:%2"
                 :: "v"(lds_byte_addr), "v"(gaddr), "i"(ioff)
                 : "memory");
}
__device__ __forceinline__ void wait_asynccnt0() {
    asm volatile("s_wait_asynccnt 0" ::: "memory");
}

// LDS layout (floats): raw copies W1 | b1 | W2 | b2
#define OFF_W1 0
#define OFF_B1 4096
#define OFF_W2 8192
#define OFF_B2 12288
#define NPAR   12544   // floats = 49.0 KB

__device__ __forceinline__ float mlp_one(float yv, const float* w1s, const float* b1s,
                                         const float* w2s, float bias2) {
    // softplus in base-2: x2 = log2e*(y*W1+b1)
    //   softplus(x) = ln2 * ( max(x2,0) + log2(1 + exp2(-|x2|)) )
    // ln2 pre-folded into w2s, so z = sum(sp2 * w2s) + b2.
    float a0 = 0.f, a1 = 0.f, a2 = 0.f, a3 = 0.f;
#pragma unroll
    for (int h = 0; h < HDIM; h += 4) {
        float x0 = __builtin_fmaf(yv, w1s[h+0], b1s[h+0]);
        float x1 = __builtin_fmaf(yv, w1s[h+1], b1s[h+1]);
        float x2 = __builtin_fmaf(yv, w1s[h+2], b1s[h+2]);
        float x3 = __builtin_fmaf(yv, w1s[h+3], b1s[h+3]);
        float u0 = __builtin_amdgcn_exp2f(-__builtin_fabsf(x0));
        float u1 = __builtin_amdgcn_exp2f(-__builtin_fabsf(x1));
        float u2 = __builtin_amdgcn_exp2f(-__builtin_fabsf(x2));
        float u3 = __builtin_amdgcn_exp2f(-__builtin_fabsf(x3));
        float s0 = __builtin_fmaxf(x0, 0.f) + __builtin_amdgcn_logf(1.f + u0);
        float s1 = __builtin_fmaxf(x1, 0.f) + __builtin_amdgcn_logf(1.f + u1);
        float s2 = __builtin_fmaxf(x2, 0.f) + __builtin_amdgcn_logf(1.f + u2);
        float s3 = __builtin_fmaxf(x3, 0.f) + __builtin_amdgcn_logf(1.f + u3);
        a0 = __builtin_fmaf(s0, w2s[h+0], a0);
        a1 = __builtin_fmaf(s1, w2s[h+1], a1);
        a2 = __builtin_fmaf(s2, w2s[h+2], a2);
        a3 = __builtin_fmaf(s3, w2s[h+3], a3);
    }
    const float z = ((a0 + a1) + (a2 + a3)) + bias2;
    // sigmoid(z) = 1 / (1 + exp2(z * -log2e))
    const float e = __builtin_amdgcn_exp2f(z * -1.4426950408889634f);
    return __builtin_amdgcn_rcpf(1.f + e);
}

__global__ __launch_bounds__(256) void diffusion_latent_mlp(
    const float* __restrict__ y,  const float* __restrict__ W1,
    const float* __restrict__ b1, const float* __restrict__ W2,
    const float* __restrict__ b2, float* __restrict__ out)
{
    __shared__ __align__(16) float s_par[NPAR];
    const int t = threadIdx.x;

    // ---- stage all parameters into LDS: one base address per array, chunk
    //      selected by the instruction IOFFSET (applies to LDS + global alike).
    {
        const unsigned base = (unsigned)(unsigned long long)(const void*)&s_par[0];
        const unsigned ldsW1 = base + (unsigned)(OFF_W1 * 4) + 16u * (unsigned)t;
        const unsigned ldsB1 = base + (unsigned)(OFF_B1 * 4) + 16u * (unsigned)t;
        const unsigned ldsW2 = base + (unsigned)(OFF_W2 * 4) + 16u * (unsigned)t;
        const float* gW1 = W1 + 4 * t;
        const float* gB1 = b1 + 4 * t;
        const float* gW2 = W2 + 4 * t;
        async_ld_b128(ldsW1, gW1, 0x0000);
        async_ld_b128(ldsW1, gW1, 0x1000);
        async_ld_b128(ldsW1, gW1, 0x2000);
        async_ld_b128(ldsW1, gW1, 0x3000);
        async_ld_b128(ldsB1, gB1, 0x0000);
        async_ld_b128(ldsB1, gB1, 0x1000);
        async_ld_b128(ldsB1, gB1, 0x2000);
        async_ld_b128(ldsB1, gB1, 0x3000);
        async_ld_b128(ldsW2, gW2, 0x0000);
        async_ld_b128(ldsW2, gW2, 0x1000);
        async_ld_b128(ldsW2, gW2, 0x2000);
        async_ld_b128(ldsW2, gW2, 0x3000);
        if (t < 64)   // b2: 64 float4
            async_ld_b128(base + (unsigned)(OFF_B2 * 4) + 16u * (unsigned)t, b2 + 4 * t, 0);
        wait_asynccnt0();
    }
    __syncthreads();

    // ---- hoist this thread's 48 weights into registers, pre-scaled for exp2/log2
    const float LOG2E = 1.4426950408889634f;  // fold into W1,b1: x2 = x*log2e
    const float LN2   = 0.6931471805599453f;  // fold into W2:    sp = ln2*sp2
    float w1s[HDIM], b1s[HDIM], w2s[HDIM];
    {
        const float4* p1 = (const float4*)(s_par + OFF_W1 + t * HDIM);
        const float4* pb = (const float4*)(s_par + OFF_B1 + t * HDIM);
        const float4* p2 = (const float4*)(s_par + OFF_W2 + t * HDIM);
#pragma unroll
        for (int i = 0; i < 4; ++i) {
            float4 a = p1[i], c = pb[i], d = p2[i];
            w1s[4*i+0] = a.x * LOG2E; w1s[4*i+1] = a.y * LOG2E;
            w1s[4*i+2] = a.z * LOG2E; w1s[4*i+3] = a.w * LOG2E;
            b1s[4*i+0] = c.x * LOG2E; b1s[4*i+1] = c.y * LOG2E;
            b1s[4*i+2] = c.z * LOG2E; b1s[4*i+3] = c.w * LOG2E;
            w2s[4*i+0] = d.x * LN2;   w2s[4*i+1] = d.y * LN2;
            w2s[4*i+2] = d.z * LN2;   w2s[4*i+3] = d.w * LN2;
        }
    }
    const float bias2 = s_par[OFF_B2 + t];

    // ---- stream batch rows: lane-contiguous along l => fully coalesced.
    //      Two rows in flight so the 128B/wave loads hide under ~140 VALU+TRANS ops.
    const float* yrow = y   + (size_t)blockIdx.x * ROWS * LDIM + t;
    float*       orow = out + (size_t)blockIdx.x * ROWS * LDIM + t;

    for (int r = 0; r < ROWS; r += 2) {
        const float y0 = yrow[(size_t)(r + 0) * LDIM];
        const float y1 = yrow[(size_t)(r + 1) * LDIM];
        if (r + 4 < ROWS) {  // global_prefetch_b8 of upcoming stream
            __builtin_prefetch(yrow + (size_t)(r + 4) * LDIM, 0, 0);
            __builtin_prefetch(yrow + (size_t)(r + 5) * LDIM, 0, 0);
        }
        const float o0 = mlp_one(y0, w1s, b1s, w2s, bias2);
        const float o1 = mlp_one(y1, w1s, b1s, w2s, bias2);
        orow[(size_t)(r + 0) * LDIM] = o0;
        orow[(size_t)(r + 1) * LDIM] = o1;
    }
}

extern "C" void kernel_launch(void* const* d_in, const int* in_sizes, int n_in,
                              void* d_out, int out_size, void* d_ws, size_t ws_size,
                              hipStream_t stream) {
    // setup_inputs order: t(0), y(1), W1(2), b1(3), W2(4), b2(5), args(6)
    const float* y  = (const float*)d_in[1];
    const float* W1 = (const float*)d_in[2];
    const float* b1 = (const float*)d_in[3];
    const float* W2 = (const float*)d_in[4];
    const float* b2 = (const float*)d_in[5];
    float* out = (float*)d_out;
    diffusion_latent_mlp<<<NBLK, 256, 0, stream>>>(y, W1, b1, W2, b2, out);
}